// NaiveMultiPartitionState_89300960019112
// MI455X (gfx1250) — compile-verified
//
#include <hip/hip_runtime.h>

// Problem constants (match reference)
#define P_    128            // NUM_PARTITIONS
#define C_    32
#define D_    128
#define B_    4
#define S_    2048
#define K_    2
#define NPAIR (B_ * S_)      // 8192 (b,s) pairs
#define NTOK  (NPAIR * K_)   // 16384 tokens

typedef __attribute__((ext_vector_type(2))) float v2f;
typedef __attribute__((ext_vector_type(8))) float v8f;

// ---------------------------------------------------------------------------
// Kernel A: ksum[t] = sum_c keys[t][c] via V_WMMA_F32_16X16X4_F32 with B=ones.
// A-matrix layout (32-bit A 16x4): lane L<16 -> M=L, {K0,K1} in {a.x,a.y};
// lane L>=16 -> M=L-16, {K2,K3}. With B all-ones the product row m is
// ksum over that K-slice, accumulated exactly in fp32 over 8 slices (C=32).
// D layout: VGPR r, lanes 0-15 => (M=r, N=lane); lanes 16-31 => (M=r+8).
// Diagonal extraction: lane m (m<8) holds c[m]=D[m,m]=ksum[tok m];
// lane 16+j holds c[j]=D[8+j,j]=ksum[tok 8+j].
// Also zeroes the global accumulator (grid is exactly P_*D_ threads).
// ---------------------------------------------------------------------------
__global__ __launch_bounds__(256) void ksum_wmma_kernel(
    const float* __restrict__ keys, float* __restrict__ ksum,
    float* __restrict__ accum) {
  const int tid = threadIdx.x;
  const int gid = blockIdx.x * 256 + tid;   // 64 blocks * 256 = 16384 = P_*D_
  accum[gid] = 0.0f;                        // re-zero scratch every call

  const int lane      = tid & 31;
  const int wid       = gid >> 5;           // global wave id: 0..511
  const int tokInLane = lane & 15;
  const int chalf     = (lane >> 4) << 1;   // 0 or 2

  v2f ones;
  ones.x = 1.0f;
  ones.y = 1.0f;

#pragma unroll
  for (int rep = 0; rep < 2; ++rep) {
    const int g       = wid + rep * 512;    // token-group 0..1023 (16 tokens each)
    const int tokbase = g * 16;
    const float* kp = keys + (size_t)(tokbase + tokInLane) * C_ + chalf;

    v8f c = {};
#pragma unroll
    for (int i = 0; i < 8; ++i) {           // 8 K-slices of 4 cover C=32
      v2f a = *(const v2f*)(kp + i * 4);    // {K0,K1} or {K2,K3} for this lane
      c = __builtin_amdgcn_wmma_f32_16x16x4_f32(
          /*neg_a=*/false, a, /*neg_b=*/false, ones,
          /*c_mod=*/(short)0, c, /*reuse_a=*/false, /*reuse_b=*/false);
    }

    // diagonal pick: r = c[lane & 7]
    const int k = lane & 7;
    float r = c[0];
#pragma unroll
    for (int i = 1; i < 8; ++i) r = (k == i) ? c[i] : r;

    if (lane < 8) {
      ksum[tokbase + lane] = r;             // tokens 0..7
    } else if (lane >= 16 && lane < 24) {
      ksum[tokbase + 8 + (lane - 16)] = r;  // tokens 8..15
    }
  }
}

// ---------------------------------------------------------------------------
// CDNA5 helpers: async global->LDS (ASYNCcnt), native DS / global fp atomics.
// ---------------------------------------------------------------------------
__device__ __forceinline__ void async_load_b32_to_lds(unsigned lds_byte_addr,
                                                      const float* gaddr) {
  asm volatile("global_load_async_to_lds_b32 %0, %1, off"
               :
               : "v"(lds_byte_addr), "v"((unsigned long long)(size_t)gaddr)
               : "memory");
}
__device__ __forceinline__ void wait_asynccnt_le1() {
  asm volatile("s_wait_asynccnt 0x1" ::: "memory");
}
__device__ __forceinline__ void wait_asynccnt_le0() {
  asm volatile("s_wait_asynccnt 0x0" ::: "memory");
}
// Single-instruction LDS float atomic add (non-returning, DScnt-tracked).
__device__ __forceinline__ void lds_atomic_add_f32(const float* lds_ptr,
                                                   float v) {
  asm volatile("ds_add_f32 %0, %1"
               :
               : "v"((unsigned)(size_t)lds_ptr), "v"(v)
               : "memory");
}
__device__ __forceinline__ void wait_dscnt_0() {
  asm volatile("s_wait_dscnt 0x0" ::: "memory");
}
// Single-instruction global float atomic add (non-returning, STOREcnt).
__device__ __forceinline__ void global_atomic_add_f32(float* gptr, float v) {
  asm volatile("global_atomic_add_f32 %0, %1, off"
               :
               : "v"((unsigned long long)(size_t)gptr), "v"(v)
               : "memory");
}

// ---------------------------------------------------------------------------
// Kernel B: scatter-accumulate accum[p][d] += ksum[t] * v[t][d].
// 32 blocks x 256 threads; each block owns 256 (b,s) pairs and a private
// 64 KB LDS accumulator (ds_add_f32, conflict-free: lanes cover contiguous d).
// Values stream is double-buffered through LDS with async-to-LDS loads.
// Block result merged into global accum via global_atomic_add_f32.
// ---------------------------------------------------------------------------
#define B_GRID   32
#define B_PAIRS  (NPAIR / B_GRID)   // 256 pairs per block
#define B_STEPS  (B_PAIRS / 2)      // 128 steps, 2 pairs per step

__global__ __launch_bounds__(256) void scatter_kernel(
    const int* __restrict__ idx, const float* __restrict__ values,
    const float* __restrict__ ksum, float* __restrict__ accum) {
  __shared__ __align__(16) float acc[P_ * D_];  // 64 KB
  __shared__ float vbuf[2][256];   // async double buffer (1 value/thread/step)

  const int tid = threadIdx.x;
  float4* acc4 = (float4*)acc;
#pragma unroll
  for (int i = tid; i < (P_ * D_) / 4; i += 256) {  // 16 x ds_store_b128
    float4 z;
    z.x = 0.0f; z.y = 0.0f; z.z = 0.0f; z.w = 0.0f;
    acc4[i] = z;
  }
  __syncthreads();

  const int pairBase = blockIdx.x * B_PAIRS;
  const int sub      = tid >> 7;    // which of the 2 pairs this step
  const int d        = tid & 127;

  // this thread's global value address advances by 2 rows (2*D_) per step
  const float* gbase = values + (size_t)(pairBase + sub) * D_ + d;
  const unsigned lds0 = (unsigned)(size_t)&vbuf[0][tid];
  const unsigned lds1 = (unsigned)(size_t)&vbuf[1][tid];

  async_load_b32_to_lds(lds0, gbase);                       // prefetch step 0

#pragma unroll 4
  for (int j = 0; j < B_STEPS - 1; ++j) {
    async_load_b32_to_lds((j & 1) ? lds0 : lds1,            // prefetch j+1
                          gbase + (size_t)(j + 1) * 2 * D_);
    wait_asynccnt_le1();                                    // step j data ready
    const int pair = pairBase + j * 2 + sub;
    const int t0   = pair << 1;
    const float ks0 = ksum[t0];
    const float ks1 = ksum[t0 + 1];
    const int   p0  = idx[t0];
    const int   p1  = idx[t0 + 1];
    const float v   = vbuf[j & 1][tid];
    lds_atomic_add_f32(&acc[p0 * D_ + d], ks0 * v);
    lds_atomic_add_f32(&acc[p1 * D_ + d], ks1 * v);
  }
  {
    wait_asynccnt_le0();                                    // last step
    const int j    = B_STEPS - 1;
    const int pair = pairBase + j * 2 + sub;
    const int t0   = pair << 1;
    const float ks0 = ksum[t0];
    const float ks1 = ksum[t0 + 1];
    const int   p0  = idx[t0];
    const int   p1  = idx[t0 + 1];
    const float v   = vbuf[j & 1][tid];
    lds_atomic_add_f32(&acc[p0 * D_ + d], ks0 * v);
    lds_atomic_add_f32(&acc[p1 * D_ + d], ks1 * v);
  }

  wait_dscnt_0();      // asm DS atomics are not in compiler's barrier bookkeeping
  __syncthreads();

#pragma unroll
  for (int i = tid; i < (P_ * D_) / 4; i += 256) {  // 16 x ds_load_b128
    float4 a = acc4[i];
    float* g = accum + 4 * i;
    global_atomic_add_f32(g + 0, a.x);
    global_atomic_add_f32(g + 1, a.y);
    global_atomic_add_f32(g + 2, a.z);
    global_atomic_add_f32(g + 3, a.w);
  }
}

// ---------------------------------------------------------------------------
// Kernel C: out[b,s,:] = (accum[p0,:] + accum[p1,:]) * (1/C) * q[b,s,:].
// One wave per pair (wave32): lane owns one float4 of D=128. All b128 traffic.
// ---------------------------------------------------------------------------
__global__ __launch_bounds__(256) void finalize_kernel(
    const int* __restrict__ idx, const float* __restrict__ queries,
    const float* __restrict__ accum, float* __restrict__ out) {
  const int tid  = threadIdx.x;
  const int lane = tid & 31;
  const int pair = (blockIdx.x * 256 + tid) >> 5;  // 1024 blocks * 8 waves = 8192
  const int t0   = pair << 1;
  const int p0 = idx[t0];
  const int p1 = idx[t0 + 1];

  const float4* acc4 = (const float4*)accum;
  const float4  m0   = acc4[p0 * (D_ / 4) + lane];
  const float4  m1   = acc4[p1 * (D_ / 4) + lane];
  const float4  q    = ((const float4*)queries)[pair * (D_ / 4) + lane];

  const float s = 1.0f / (float)C_;
  float4 o;
  o.x = (m0.x + m1.x) * s * q.x;
  o.y = (m0.y + m1.y) * s * q.y;
  o.z = (m0.z + m1.z) * s * q.z;
  o.w = (m0.w + m1.w) * s * q.w;
  ((float4*)out)[pair * (D_ / 4) + lane] = o;
}

// ---------------------------------------------------------------------------
extern "C" void kernel_launch(void* const* d_in, const int* in_sizes, int n_in,
                              void* d_out, int out_size, void* d_ws,
                              size_t ws_size, hipStream_t stream) {
  const int*   idx  = (const int*)d_in[0];    // [B,S,K]
  const float* keys = (const float*)d_in[1];  // [B,S,K,C]
  const float* vals = (const float*)d_in[2];  // [B,S,D]
  const float* qry  = (const float*)d_in[3];  // [B,S,D]
  float*       out  = (float*)d_out;          // [B,S,D]

  // workspace layout: ksum (16384 f32) | accum (16384 f32)  => 128 KB total
  float* ksum  = (float*)d_ws;
  float* accum = ksum + NTOK;

  hipLaunchKernelGGL(ksum_wmma_kernel, dim3(64), dim3(256), 0, stream,
                     keys, ksum, accum);
  hipLaunchKernelGGL(scatter_kernel, dim3(B_GRID), dim3(256), 0, stream,
                     idx, vals, ksum, accum);
  hipLaunchKernelGGL(finalize_kernel, dim3(NPAIR / 8), dim3(256), 0, stream,
                     idx, qry, accum, out);
}